// PillarFusionFeatureNet_64029372449486
// MI455X (gfx1250) — compile-verified
//
#include <hip/hip_runtime.h>
#include <math.h>

// ---------------------------------------------------------------------------
// PillarFusionFeatureNet for MI455X (gfx1250)
//   - HBM-bound streaming kernel (~83 MB => ~3.6us floor @ 23.3 TB/s)
//   - CDNA5 async data path: global_load_async_to_lds_b64 + s_wait_asynccnt
//   - 2 threads per pillar, conflict-free LDS slab stride (1288 B == 322 words,
//     322 % 64 == 2 -> ds_load_b64 starting banks all distinct across a wave)
//   - no WMMA: arithmetic intensity ~1 FLOP/byte, orders of magnitude below
//     any matrix-core ceiling; the win is the data-movement path.
// ---------------------------------------------------------------------------

namespace {
constexpr int kNPillars        = 60000;
constexpr int kMaxPts          = 32;
constexpr int kC               = 10;
constexpr int kPillarBytes     = kMaxPts * kC * 4;     // 1280
constexpr int kSlabBytes       = kPillarBytes + 8;     // 1288: bank shift of 2 words/slab
constexpr int kSlabWords       = kSlabBytes / 4;       // 322
constexpr int kPillarsPerBlock = 32;                   // 2 lanes per pillar
constexpr int kThreads         = 64;                   // 2 waves
constexpr int kStageIters      = (kPillarsPerBlock * kPillarBytes / 8) / kThreads; // 80
constexpr float kVX = 0.2f, kVY = 0.2f, kVZ = 4.0f;
constexpr float kXOff = 0.1f, kYOff = -39.9f, kZOff = -1.0f;

// Every block is full: staging loop can be branch-free (no exec masking).
static_assert(kNPillars % kPillarsPerBlock == 0, "grid exactly covers pillars");
}  // namespace

__global__ __launch_bounds__(kThreads)
void pillar_fusion_kernel(const float* __restrict__ features,
                          const int*   __restrict__ num_points,
                          const int*   __restrict__ coors,
                          const float* __restrict__ W_proj,
                          float*       __restrict__ out,
                          int n_pillars)
{
    __shared__ float smem[kPillarsPerBlock * kSlabWords];   // 41216 B

    const int t          = threadIdx.x;                     // 0..63
    const int block_base = blockIdx.x * kPillarsPerBlock;

    // ---------------- stage 32 pillars (40 KB) into LDS asynchronously -----
    {
        const unsigned long long gbase =
            (unsigned long long)(uintptr_t)features +
            (unsigned long long)block_base * (unsigned long long)kPillarBytes;
        const unsigned lds_base = (unsigned)(uintptr_t)&smem[0];

        // 8-byte chunks; consecutive lanes -> consecutive global addresses
        // (fully coalesced 256B bursts per wave instruction).
#pragma unroll 4
        for (int i = 0; i < kStageIters; ++i) {
            unsigned goff = (unsigned)(i * kThreads + t) * 8u;
            unsigned j    = goff / (unsigned)kPillarBytes;     // pillar in block
            unsigned lds  = lds_base + goff + j * 8u;          // padded slab layout
            asm volatile("global_load_async_to_lds_b64 %0, %1, %2"
                         :: "v"(lds), "v"(goff), "s"(gbase)
                         : "memory");
        }
        asm volatile("s_wait_asynccnt 0" ::: "memory");
    }
    __syncthreads();   // other wave waited on its own ASYNCcnt before arriving

    // ---------------- compute: lane pair (half=0/1) per pillar -------------
    const int j    = t >> 1;          // pillar within block
    const int half = t & 1;           // which 16 points
    const int p    = block_base + j;
    if (p >= n_pillars) return;       // never taken; cheap safety net

    const float* slab = smem + j * kSlabWords + half * (16 * kC);

    float sx = 0.f, sy = 0.f;
    float rc = 0.f, r5 = 0.f, r6 = 0.f, r7 = 0.f, r8 = 0.f;
    float lc = 0.f, l0 = 0.f, l1 = 0.f, l2 = 0.f, l3 = 0.f, l4 = 0.f;
    float cnt = 0.f, ssq = 0.f;
    float h[16];

#pragma unroll
    for (int i = 0; i < 16; ++i) {
        const float* fp = slab + i * kC;
        float2 a  = *(const float2*)(fp + 0);   // f0 f1
        float2 b  = *(const float2*)(fp + 2);   // f2 f3
        float2 c2 = *(const float2*)(fp + 4);   // f4 f5
        float2 d  = *(const float2*)(fp + 6);   // f6 f7
        float2 e  = *(const float2*)(fp + 8);   // f8 flag

        float flag = e.y;
        float mr = (flag ==  1.0f) ? 1.0f : 0.0f;
        float ml = (flag == -1.0f) ? 1.0f : 0.0f;

        sx += a.x;  sy += a.y;
        rc += mr;   r5 += mr * c2.y; r6 += mr * d.x; r7 += mr * d.y; r8 += mr * e.x;
        lc += ml;   l0 += ml * a.x;  l1 += ml * a.y; l2 += ml * b.x;
        l3 += ml * b.y; l4 += ml * c2.x;

        float hh = b.x * ml;            // heights = f2 * mask_lidar
        h[i] = hh;
        cnt += (hh != 0.0f) ? 1.0f : 0.0f;
        ssq += hh * hh;
    }

    // combine the two halves of the pillar (lanes 2j and 2j+1)
#define CDNA5_PAIR_SUM(x) x += __shfl_xor(x, 1, 32)
    CDNA5_PAIR_SUM(sx);  CDNA5_PAIR_SUM(sy);
    CDNA5_PAIR_SUM(rc);  CDNA5_PAIR_SUM(r5); CDNA5_PAIR_SUM(r6);
    CDNA5_PAIR_SUM(r7);  CDNA5_PAIR_SUM(r8);
    CDNA5_PAIR_SUM(lc);  CDNA5_PAIR_SUM(l0); CDNA5_PAIR_SUM(l1);
    CDNA5_PAIR_SUM(l2);  CDNA5_PAIR_SUM(l3); CDNA5_PAIR_SUM(l4);
    CDNA5_PAIR_SUM(cnt); CDNA5_PAIR_SUM(ssq);

    const float s = l2;                              // sum(heights) == lidar sum ch2

    // Gaussian-weighted height statistics
    float cnt_c  = fmaxf(cnt, 1.0f);
    float mean   = s / cnt_c;
    float varnum = ssq - 2.0f * mean * s + cnt * mean * mean;
    float var    = fmaxf(varnum, 0.0f) / fmaxf(cnt - 1.0f, 1.0f);
    float stdv   = sqrtf(var);
    float sstd   = (stdv == 0.0f) ? 1.0f : stdv;
    float inv_ss = 1.0f / sstd;

    float wsum = 0.f, hw = 0.f;
#pragma unroll
    for (int i = 0; i < 16; ++i) {
        float hh = h[i];
        float vm = (hh != 0.0f) ? 1.0f : 0.0f;
        float z  = (hh - mean) * inv_ss;
        float w  = __expf(-0.5f * z * z) * vm;
        wsum += w;
        hw   += hh * w;
    }
    CDNA5_PAIR_SUM(wsum);
    CDNA5_PAIR_SUM(hw);
#undef CDNA5_PAIR_SUM

    float weighted = hw / fmaxf(wsum, 1e-12f);
    float out_h = (cnt == 0.0f) ? 0.0f
                : (cnt == 1.0f) ? s
                : (stdv == 0.0f) ? mean
                : weighted;

    if (half == 0) {
        float inv_np = 1.0f / (float)num_points[p];
        float ox = sx * inv_np, oy = sy * inv_np;

        float rcc = fmaxf(rc, 1.0f);
        float or0 = r5 / rcc, or1 = r6 / rcc, or2 = r7 / rcc, or3 = r8 / rcc;

        float lcc = fmaxf(lc, 1.0f);
        float lm0 = l0 / lcc, lm1 = l1 / lcc, lm2 = l2 / lcc;
        float p3  = l3 / lcc, p4  = l4 / lcc;

        float w00 = W_proj[0], w01 = W_proj[1], w10 = W_proj[2], w11 = W_proj[3];
        float op0 = p3 * w00 + p4 * w01;    // lidar_35 @ W_proj.T
        float op1 = p3 * w10 + p4 * w11;

        bool hasl = lc > 0.0f;
        float oc0 = hasl ? (lm0 - ox)    : 0.0f;
        float oc1 = hasl ? (lm1 - oy)    : 0.0f;
        float oc2 = hasl ? (lm2 - out_h) : 0.0f;

        int c1 = coors[p * 4 + 1], c2i = coors[p * 4 + 2], c3 = coors[p * 4 + 3];
        float cx = (float)c3  * kVX + kXOff;
        float cy = (float)c2i * kVY + kYOff;
        float cz = (float)c1  * kVZ + kZOff;
        float ce0 = hasl ? (lm0 - cx) : 0.0f;
        float ce1 = hasl ? (lm1 - cy) : 0.0f;
        float ce2 = hasl ? (lm2 - cz) : 0.0f;

        float4* orow = (float4*)(out + (size_t)p * 16);
        orow[0] = make_float4(ox,  oy,  out_h, op0);
        orow[1] = make_float4(op1, or0, or1,   or2);
        orow[2] = make_float4(or3, oc0, oc1,   oc2);
        orow[3] = make_float4(ce0, ce1, ce2,   0.0f);
    }
}

extern "C" void kernel_launch(void* const* d_in, const int* in_sizes, int n_in,
                              void* d_out, int out_size, void* d_ws, size_t ws_size,
                              hipStream_t stream) {
    (void)in_sizes; (void)n_in; (void)out_size; (void)d_ws; (void)ws_size;

    const float* features   = (const float*)d_in[0];
    const int*   num_points = (const int*)  d_in[1];
    const int*   coors      = (const int*)  d_in[2];
    const float* W_proj     = (const float*)d_in[3];
    float*       out        = (float*)      d_out;

    const int blocks = kNPillars / kPillarsPerBlock;   // 1875, exact cover
    pillar_fusion_kernel<<<blocks, kThreads, 0, stream>>>(
        features, num_points, coors, W_proj, out, kNPillars);

    // Tuple tail: raw pass-through of num_points then coors after the 60000x16 out.
    char* tail = (char*)d_out + (size_t)kNPillars * 16 * sizeof(float);
    hipMemcpyAsync(tail, d_in[1], (size_t)kNPillars * sizeof(int),
                   hipMemcpyDeviceToDevice, stream);
    hipMemcpyAsync(tail + (size_t)kNPillars * sizeof(int), d_in[2],
                   (size_t)kNPillars * 4 * sizeof(int),
                   hipMemcpyDeviceToDevice, stream);
}